// Transformer_21191368638576
// MI455X (gfx1250) — compile-verified
//
#include <hip/hip_runtime.h>

typedef float v2f __attribute__((ext_vector_type(2)));
typedef float v8f __attribute__((ext_vector_type(8)));

#define NTGT 1024
#define NMEM 2048
#define BATCH 2
#define DMODEL 512
#define NHEAD 8
#define HEADD 64
#define DFF 2048
#define LLAYERS 4

static __device__ __forceinline__ v8f wmma_f32(v2f a, v2f b, v8f c) {
  // D = A(16x4) * B(4x16) + C(16x16), fp32
  return __builtin_amdgcn_wmma_f32_16x16x4_f32(false, a, false, b, (short)0, c,
                                               false, false);
}

// ---------------------------------------------------------------------------
// Elementwise add: out = a + b
// ---------------------------------------------------------------------------
__global__ void add_kernel(const float* __restrict__ a,
                           const float* __restrict__ b,
                           float* __restrict__ out, int n) {
  int i = blockIdx.x * blockDim.x + threadIdx.x;
  if (i < n) out[i] = a[i] + b[i];
}

// ---------------------------------------------------------------------------
// Tiled WMMA GEMM: C[M,N] = A[M,K] @ op(B) + bias, optional ReLU.
// transB=1: B is W[N,K] row-major (PyTorch Linear weight), C = A @ W^T.
// transB=0: B is [K,N] row-major.
// Block tile 128x64, 8 waves, each wave a 32x32 patch (2x2 WMMA tiles).
// Requires M%128==0, N%64==0, K%16==0 (true for all uses here).
// ---------------------------------------------------------------------------
#define BM 128
#define BN 64
#define BK 16

__global__ __launch_bounds__(256) void gemm_wmma_kernel(
    const float* __restrict__ A, int lda, const float* __restrict__ Bm, int ldb,
    float* __restrict__ C, int ldc, const float* __restrict__ bias, int M,
    int N, int K, int transB, int relu) {
  __shared__ float As[BK][BM];
  __shared__ float Bs[BK][BN];

  const int tid = threadIdx.x;
  const int lane = tid & 31;
  const int w = tid >> 5;       // 0..7
  const int wm = w >> 1;        // 0..3
  const int wn = w & 1;         // 0..1
  const int m0 = blockIdx.x * BM;
  const int n0 = blockIdx.y * BN;
  const int row = lane & 15;
  const int c0 = (lane < 16) ? 0 : 2;

  v8f acc[2][2];
#pragma unroll
  for (int i = 0; i < 2; i++)
#pragma unroll
    for (int j = 0; j < 2; j++)
#pragma unroll
      for (int t = 0; t < 8; t++) acc[i][j][t] = 0.0f;

  for (int k0 = 0; k0 < K; k0 += BK) {
    // Stage A tile (128x16), transposed into As[k][m]
    {
      int m = tid >> 1;
      int off = (tid & 1) * 8;
      const float* ap = A + (size_t)(m0 + m) * lda + k0 + off;
#pragma unroll
      for (int j = 0; j < 8; j++) As[off + j][m] = ap[j];
    }
    // Stage B tile (16x64) as Bs[k][n]
    if (transB) {
      for (int idx = tid; idx < BK * BN; idx += 256) {
        int kk = idx & 15, n = idx >> 4;
        Bs[kk][n] = Bm[(size_t)(n0 + n) * ldb + k0 + kk];
      }
    } else {
      for (int idx = tid; idx < BK * BN; idx += 256) {
        int n = idx & 63, kk = idx >> 6;
        Bs[kk][n] = Bm[(size_t)(k0 + kk) * ldb + n0 + n];
      }
    }
    __syncthreads();

#pragma unroll
    for (int ks = 0; ks < 4; ks++) {
      const int kb = ks * 4 + c0;
      v2f af[2], bf[2];
#pragma unroll
      for (int i = 0; i < 2; i++) {
        af[i].x = As[kb][wm * 32 + i * 16 + row];
        af[i].y = As[kb + 1][wm * 32 + i * 16 + row];
      }
#pragma unroll
      for (int j = 0; j < 2; j++) {
        bf[j].x = Bs[kb][wn * 32 + j * 16 + row];
        bf[j].y = Bs[kb + 1][wn * 32 + j * 16 + row];
      }
#pragma unroll
      for (int i = 0; i < 2; i++)
#pragma unroll
        for (int j = 0; j < 2; j++)
          acc[i][j] = wmma_f32(af[i], bf[j], acc[i][j]);
    }
    __syncthreads();
  }

  const int rhi = (lane >> 4) * 8;
#pragma unroll
  for (int i = 0; i < 2; i++) {
#pragma unroll
    for (int j = 0; j < 2; j++) {
      int cc = n0 + wn * 32 + j * 16 + row;
      float bv = bias ? bias[cc] : 0.0f;
#pragma unroll
      for (int t = 0; t < 8; t++) {
        int rr = m0 + wm * 32 + i * 16 + t + rhi;
        float val = acc[i][j][t] + bv;
        if (relu) val = fmaxf(val, 0.0f);
        C[(size_t)rr * ldc + cc] = val;
      }
    }
  }
}

// ---------------------------------------------------------------------------
// Flash self-attention (fp32 WMMA). One wave per 16 query rows per (b,h).
// qk: [2048 rows (n*2+b), 1024] cols 0..511 = Q proj, 512..1023 = K proj.
// v : [2048 rows, 512].  o : [2048 rows, 512].  scale = 1/8.
// ---------------------------------------------------------------------------
__global__ __launch_bounds__(32) void flash_attn_kernel(
    const float* __restrict__ qk, const float* __restrict__ v,
    float* __restrict__ o) {
  const int qb = blockIdx.x;           // query block 0..63
  const int bh = blockIdx.y;           // b*8+h
  const int b = bh >> 3, h = bh & 7;

  const float* Q = qk + (size_t)b * 1024 + h * 64;        // row stride 2048
  const float* Kp = qk + (size_t)b * 1024 + 512 + h * 64; // row stride 2048
  const float* Vp = v + (size_t)b * 512 + h * 64;         // row stride 1024
  float* O = o + (size_t)b * 512 + h * 64;                // row stride 1024

  __shared__ float kT[16][68];
  __shared__ float vT[16][68];
  __shared__ float sT[16][17];
  __shared__ float pT[16][17];
  __shared__ float mrow[16], lrow[16], frow[16];

  const int lane = threadIdx.x;
  const int row = lane & 15;
  const int c0 = (lane < 16) ? 0 : 2;
  const int hi = (lane >> 4) * 8;

  // Preload this lane's Q operand fragments (its half of the row's K-slices)
  float qa[32];
  {
    const float* qp = Q + (size_t)(qb * 16 + row) * 2048;
#pragma unroll
    for (int k = 0; k < 16; k++) {
      qa[2 * k] = qp[4 * k + c0];
      qa[2 * k + 1] = qp[4 * k + c0 + 1];
    }
  }
  if (lane < 16) {
    mrow[lane] = -1e30f;
    lrow[lane] = 0.0f;
  }

  v8f oacc[4];
#pragma unroll
  for (int c = 0; c < 4; c++)
#pragma unroll
    for (int t = 0; t < 8; t++) oacc[c][t] = 0.0f;
  __syncthreads();

  for (int m0 = 0; m0 < NTGT; m0 += 16) {
    // Stage K,V tiles (16 keys x 64 dims), coalesced
    for (int idx = lane; idx < 16 * 64; idx += 32) {
      int r = idx >> 6, cidx = idx & 63;
      kT[r][cidx] = Kp[(size_t)(m0 + r) * 2048 + cidx];
      vT[r][cidx] = Vp[(size_t)(m0 + r) * 1024 + cidx];
    }
    __syncthreads();

    // S = Q (16x64) x K^T (64x16)
    v8f s;
#pragma unroll
    for (int t = 0; t < 8; t++) s[t] = 0.0f;
#pragma unroll
    for (int k = 0; k < 16; k++) {
      v2f a, bb;
      a.x = qa[2 * k];
      a.y = qa[2 * k + 1];
      bb.x = kT[row][4 * k + c0];      // row == column (key index) here
      bb.y = kT[row][4 * k + c0 + 1];
      s = wmma_f32(a, bb, s);
    }
#pragma unroll
    for (int t = 0; t < 8; t++) sT[t + hi][row] = s[t] * 0.125f;
    __syncthreads();

    // Online softmax update, one row per lane (lanes 0..15)
    if (lane < 16) {
      float mold = mrow[lane];
      float mx = mold;
#pragma unroll
      for (int j = 0; j < 16; j++) mx = fmaxf(mx, sT[lane][j]);
      float f = expf(mold - mx);
      float sum = lrow[lane] * f;
#pragma unroll
      for (int j = 0; j < 16; j++) {
        float p = expf(sT[lane][j] - mx);
        pT[lane][j] = p;
        sum += p;
      }
      mrow[lane] = mx;
      lrow[lane] = sum;
      frow[lane] = f;
    }
    __syncthreads();

    // Rescale accumulators by exp(m_old - m_new)
#pragma unroll
    for (int c = 0; c < 4; c++)
#pragma unroll
      for (int t = 0; t < 8; t++) oacc[c][t] *= frow[t + hi];

    // O += P (16x16) x V (16x64)
#pragma unroll
    for (int k2 = 0; k2 < 4; k2++) {
      v2f a;
      a.x = pT[row][4 * k2 + c0];
      a.y = pT[row][4 * k2 + c0 + 1];
#pragma unroll
      for (int c = 0; c < 4; c++) {
        v2f bb;
        bb.x = vT[4 * k2 + c0][c * 16 + row];
        bb.y = vT[4 * k2 + c0 + 1][c * 16 + row];
        oacc[c] = wmma_f32(a, bb, oacc[c]);
      }
    }
    __syncthreads();
  }

  // Normalize and store
#pragma unroll
  for (int t = 0; t < 8; t++) {
    float inv = 1.0f / lrow[t + hi];
    int rr = qb * 16 + t + hi;
#pragma unroll
    for (int c = 0; c < 4; c++)
      O[(size_t)rr * 1024 + c * 16 + row] = oacc[c][t] * inv;
  }
}

// ---------------------------------------------------------------------------
// Residual-add + LayerNorm over D=512. mode: 0=per-row t2, 1=broadcast over
// n (t2 row = batch index), 2 = no add (final LN). One block per row.
// ---------------------------------------------------------------------------
__global__ __launch_bounds__(256) void addln_kernel(
    const float* __restrict__ xin, const float* __restrict__ t2, int mode,
    const float* __restrict__ g, const float* __restrict__ beta,
    float* __restrict__ out) {
  int r = blockIdx.x;  // 0..2047, r = n*2 + b
  int tid = threadIdx.x;
  __shared__ float red[256];
  __shared__ float red2[256];

  const float* xr = xin + (size_t)r * DMODEL;
  const float* tr = nullptr;
  if (mode == 0) tr = t2 + (size_t)r * DMODEL;
  else if (mode == 1) tr = t2 + (size_t)(r & 1) * DMODEL;

  float v0 = xr[tid] + (tr ? tr[tid] : 0.0f);
  float v1 = xr[tid + 256] + (tr ? tr[tid + 256] : 0.0f);
  red[tid] = v0 + v1;
  red2[tid] = v0 * v0 + v1 * v1;
  __syncthreads();
  for (int off = 128; off > 0; off >>= 1) {
    if (tid < off) {
      red[tid] += red[tid + off];
      red2[tid] += red2[tid + off];
    }
    __syncthreads();
  }
  float mean = red[0] * (1.0f / DMODEL);
  float var = red2[0] * (1.0f / DMODEL) - mean * mean;
  float inv = rsqrtf(var + 1e-5f);
  float* orow = out + (size_t)r * DMODEL;
  orow[tid] = (v0 - mean) * inv * g[tid] + beta[tid];
  orow[tid + 256] = (v1 - mean) * inv * g[tid + 256] + beta[tid + 256];
}

// ---------------------------------------------------------------------------
// Custom attention helpers (cheap reductions, <1% of FLOPs)
// ---------------------------------------------------------------------------
// qsum[b*512+o] = sum_n qcp[(n*2+b)*512 + o]
__global__ void qsum_kernel(const float* __restrict__ qcp,
                            float* __restrict__ qsum) {
  int idx = blockIdx.x * blockDim.x + threadIdx.x;
  if (idx >= BATCH * DMODEL) return;
  int b = idx >> 9, o = idx & 511;
  float s = 0.0f;
  for (int n = 0; n < NTGT; n++) s += qcp[(size_t)(n * 2 + b) * DMODEL + o];
  qsum[idx] = s;
}

// s1[(b*8+h)*2048+m] = (1/8) * sum_hd qsum[b*512 + hd*8 + h] * kp[(m*2+b)*512 + hd*8+h]
__global__ void s1_kernel(const float* __restrict__ qsum,
                          const float* __restrict__ kp,
                          float* __restrict__ s1) {
  int idx = blockIdx.x * blockDim.x + threadIdx.x;
  if (idx >= BATCH * NHEAD * NMEM) return;
  int m = idx & 2047;
  int bh = idx >> 11;
  int b = bh >> 3, h = bh & 7;
  const float* kr = kp + (size_t)(m * 2 + b) * DMODEL + h;
  const float* qs = qsum + b * DMODEL + h;
  float s = 0.0f;
  for (int hd = 0; hd < HEADD; hd++) s += qs[hd * 8] * kr[hd * 8];
  s1[idx] = s * 0.125f;
}

// prob[(b*512+o)*2048+m] = softmax_m( kp[(m*2+b)*512+o] * s1[(b*8+o%8)*2048+m] )
__global__ __launch_bounds__(256) void s2softmax_kernel(
    const float* __restrict__ kp, const float* __restrict__ s1,
    float* __restrict__ prob) {
  int bo = blockIdx.x;
  int b = bo >> 9, oc = bo & 511;
  int h = oc & 7;
  int tid = threadIdx.x;
  const float* s1r = s1 + (size_t)(b * 8 + h) * NMEM;
  __shared__ float red[256];

  float mx = -1e30f;
  for (int m = tid; m < NMEM; m += 256) {
    float vv = kp[(size_t)(m * 2 + b) * DMODEL + oc] * s1r[m];
    mx = fmaxf(mx, vv);
  }
  red[tid] = mx;
  __syncthreads();
  for (int off = 128; off > 0; off >>= 1) {
    if (tid < off) red[tid] = fmaxf(red[tid], red[tid + off]);
    __syncthreads();
  }
  mx = red[0];
  __syncthreads();

  float sum = 0.0f;
  for (int m = tid; m < NMEM; m += 256) {
    float vv = kp[(size_t)(m * 2 + b) * DMODEL + oc] * s1r[m];
    sum += expf(vv - mx);
  }
  red[tid] = sum;
  __syncthreads();
  for (int off = 128; off > 0; off >>= 1) {
    if (tid < off) red[tid] += red[tid + off];
    __syncthreads();
  }
  float inv = 1.0f / red[0];
  for (int m = tid; m < NMEM; m += 256) {
    float vv = kp[(size_t)(m * 2 + b) * DMODEL + oc] * s1r[m];
    prob[(size_t)bo * NMEM + m] = expf(vv - mx) * inv;
  }
}

// w2[(b*8+h)*2048+m] = sum_hd mw[hd] * prob[(b*512 + hd*8 + h)*2048 + m]
__global__ void w2_kernel(const float* __restrict__ prob,
                          const float* __restrict__ mw,
                          float* __restrict__ w2) {
  int idx = blockIdx.x * blockDim.x + threadIdx.x;
  if (idx >= BATCH * NHEAD * NMEM) return;
  int m = idx & 2047;
  int bh = idx >> 11;
  int b = bh >> 3, h = bh & 7;
  float s = 0.0f;
  for (int hd = 0; hd < HEADD; hd++)
    s += mw[hd] * prob[(size_t)(b * DMODEL + hd * 8 + h) * NMEM + m];
  w2[idx] = s;
}

// t2c[b*512+o] = mb + sum_m w2[(b*8+o%8)*2048+m] * vp[(m*2+b)*512 + o]
__global__ __launch_bounds__(256) void t2c_kernel(const float* __restrict__ w2,
                                                  const float* __restrict__ vp,
                                                  const float* __restrict__ mbp,
                                                  float* __restrict__ t2c) {
  int bo = blockIdx.x;
  int b = bo >> 9, oc = bo & 511;
  int h = oc & 7;
  int tid = threadIdx.x;
  const float* w2r = w2 + (size_t)(b * 8 + h) * NMEM;
  __shared__ float red[256];
  float s = 0.0f;
  for (int m = tid; m < NMEM; m += 256)
    s += w2r[m] * vp[(size_t)(m * 2 + b) * DMODEL + oc];
  red[tid] = s;
  __syncthreads();
  for (int off = 128; off > 0; off >>= 1) {
    if (tid < off) red[tid] += red[tid + off];
    __syncthreads();
  }
  if (tid == 0) t2c[bo] = red[0] + mbp[0];
}

// ---------------------------------------------------------------------------
// Host driver
// ---------------------------------------------------------------------------
extern "C" void kernel_launch(void* const* d_in, const int* in_sizes, int n_in,
                              void* d_out, int out_size, void* d_ws,
                              size_t ws_size, hipStream_t stream) {
  (void)in_sizes; (void)n_in; (void)out_size; (void)ws_size;

  const float* tgt = (const float*)d_in[0];
  const float* memory = (const float*)d_in[1];
  const float* query_pos = (const float*)d_in[2];
  const float* pos = (const float*)d_in[3];
  const float* ipw_all = (const float*)d_in[4];
  const float* ipb_all = (const float*)d_in[5];
  const float* ow_all = (const float*)d_in[6];
  const float* ob_all = (const float*)d_in[7];
  const float* cw_all = (const float*)d_in[8];
  const float* cb_all = (const float*)d_in[9];
  const float* mw_all = (const float*)d_in[10];
  const float* mb_all = (const float*)d_in[11];
  const float* l1w_all = (const float*)d_in[12];
  const float* l1b_all = (const float*)d_in[13];
  const float* l2w_all = (const float*)d_in[14];
  const float* l2b_all = (const float*)d_in[15];
  const float* lng_all = (const float*)d_in[16];
  const float* lnb_all = (const float*)d_in[17];
  const float* final_g = (const float*)d_in[18];
  const float* final_b = (const float*)d_in[19];

  // Workspace carve-up (floats)
  float* ws = (float*)d_ws;
  size_t off = 0;
  auto alloc = [&](size_t n) { float* p = ws + off; off += n; return p; };
  float* x = alloc((size_t)NTGT * BATCH * DMODEL);     // current tgt
  float* qadd = alloc((size_t)NTGT * BATCH * DMODEL);  // tgt + query_pos
  float* qkb = alloc((size_t)NTGT * BATCH * 1024);     // [2048,1024] q|k proj
  float* vbuf = alloc((size_t)NTGT * BATCH * DMODEL);
  float* attn = alloc((size_t)NTGT * BATCH * DMODEL);
  float* t2 = alloc((size_t)NTGT * BATCH * DMODEL);
  float* km = alloc((size_t)NMEM * BATCH * DMODEL);    // memory + pos
  float* qcp = alloc((size_t)NTGT * BATCH * DMODEL);
  float* kp = alloc((size_t)NMEM * BATCH * DMODEL);
  float* vp = alloc((size_t)NMEM * BATCH * DMODEL);
  float* qsum = alloc(BATCH * DMODEL);
  float* s1 = alloc((size_t)BATCH * NHEAD * NMEM);
  float* prob = alloc((size_t)BATCH * DMODEL * NMEM);
  float* w2 = alloc((size_t)BATCH * NHEAD * NMEM);
  float* t2c = alloc(BATCH * DMODEL);
  float* hbuf = alloc((size_t)NTGT * BATCH * DFF);

  auto gemm = [&](const float* A, int lda, const float* Bm, int ldb, float* C,
                  int ldc, const float* bias, int M, int N, int K, int transB,
                  int relu) {
    dim3 grid(M / BM, N / BN);
    gemm_wmma_kernel<<<grid, dim3(256), 0, stream>>>(A, lda, Bm, ldb, C, ldc,
                                                     bias, M, N, K, transB,
                                                     relu);
  };

  const int NB = NTGT * BATCH;   // 2048 rows
  const int MB = NMEM * BATCH;   // 4096 rows

  // x = tgt
  hipMemcpyAsync(x, tgt, (size_t)NB * DMODEL * sizeof(float),
                 hipMemcpyDeviceToDevice, stream);
  // km = memory + pos (layer-invariant)
  {
    int n = MB * DMODEL;
    add_kernel<<<(n + 255) / 256, 256, 0, stream>>>(memory, pos, km, n);
  }

  for (int i = 0; i < LLAYERS; i++) {
    const float* ipw = ipw_all + (size_t)i * 3 * DMODEL * DMODEL;
    const float* ipb = ipb_all + (size_t)i * 3 * DMODEL;
    const float* owp = ow_all + (size_t)i * DMODEL * DMODEL;
    const float* obp = ob_all + (size_t)i * DMODEL;
    const float* cw = cw_all + (size_t)i * 3 * DMODEL * DMODEL;
    const float* cb = cb_all + (size_t)i * 3 * DMODEL;
    const float* mwp = mw_all + (size_t)i * HEADD;
    const float* mbp = mb_all + (size_t)i;
    const float* l1w = l1w_all + (size_t)i * DFF * DMODEL;
    const float* l1b = l1b_all + (size_t)i * DFF;
    const float* l2w = l2w_all + (size_t)i * DMODEL * DFF;
    const float* l2b = l2b_all + (size_t)i * DMODEL;
    const float* lng = lng_all + (size_t)i * 3 * DMODEL;
    const float* lnb = lnb_all + (size_t)i * 3 * DMODEL;

    // ---- Self attention ----
    {
      int n = NB * DMODEL;
      add_kernel<<<(n + 255) / 256, 256, 0, stream>>>(x, query_pos, qadd, n);
    }
    // Q|K projection (wq;wk rows of in_proj), V projection from x
    gemm(qadd, DMODEL, ipw, DMODEL, qkb, 1024, ipb, NB, 1024, DMODEL, 1, 0);
    gemm(x, DMODEL, ipw + (size_t)1024 * DMODEL, DMODEL, vbuf, DMODEL,
         ipb + 1024, NB, DMODEL, DMODEL, 1, 0);
    // Flash attention over all (b,h)
    flash_attn_kernel<<<dim3(NTGT / 16, BATCH * NHEAD), dim3(32), 0, stream>>>(
        qkb, vbuf, attn);
    // Out projection
    gemm(attn, DMODEL, owp, DMODEL, t2, DMODEL, obp, NB, DMODEL, DMODEL, 1, 0);
    addln_kernel<<<NB, 256, 0, stream>>>(x, t2, 0, lng, lnb, x);

    // ---- Custom key-modulated attention ----
    {
      int n = NB * DMODEL;
      add_kernel<<<(n + 255) / 256, 256, 0, stream>>>(x, query_pos, qadd, n);
    }
    gemm(qadd, DMODEL, cw, DMODEL, qcp, DMODEL, cb, NB, DMODEL, DMODEL, 1, 0);
    gemm(km, DMODEL, cw + (size_t)DMODEL * DMODEL, DMODEL, kp, DMODEL,
         cb + DMODEL, MB, DMODEL, DMODEL, 1, 0);
    gemm(memory, DMODEL, cw + (size_t)2 * DMODEL * DMODEL, DMODEL, vp, DMODEL,
         cb + 2 * DMODEL, MB, DMODEL, DMODEL, 1, 0);
    qsum_kernel<<<(BATCH * DMODEL + 255) / 256, 256, 0, stream>>>(qcp, qsum);
    s1_kernel<<<(BATCH * NHEAD * NMEM + 255) / 256, 256, 0, stream>>>(qsum, kp,
                                                                      s1);
    s2softmax_kernel<<<BATCH * DMODEL, 256, 0, stream>>>(kp, s1, prob);
    w2_kernel<<<(BATCH * NHEAD * NMEM + 255) / 256, 256, 0, stream>>>(prob, mwp,
                                                                      w2);
    t2c_kernel<<<BATCH * DMODEL, 256, 0, stream>>>(w2, vp, mbp, t2c);
    addln_kernel<<<NB, 256, 0, stream>>>(x, t2c, 1, lng + DMODEL, lnb + DMODEL,
                                         x);

    // ---- FFN ----
    gemm(x, DMODEL, l1w, DMODEL, hbuf, DFF, l1b, NB, DFF, DMODEL, 1, 1);
    gemm(hbuf, DFF, l2w, DFF, t2, DMODEL, l2b, NB, DMODEL, DFF, 1, 0);
    addln_kernel<<<NB, 256, 0, stream>>>(x, t2, 0, lng + 2 * DMODEL,
                                         lnb + 2 * DMODEL, x);
  }

  // Final LayerNorm straight into d_out ([1, N, B, D] == flat [N*B*D])
  addln_kernel<<<NB, 256, 0, stream>>>(x, nullptr, 2, final_g, final_b,
                                       (float*)d_out);
}